// DiffusionLoss_2370821947571
// MI455X (gfx1250) — compile-verified
//
#include <hip/hip_runtime.h>

typedef float v2f __attribute__((ext_vector_type(2)));
typedef float v8f __attribute__((ext_vector_type(8)));

#define LTOT 4096
#define DD   4
#define NT   256                 // 16x16 tiles per side (LTOT/16)
#define TT   (NT*(NT+1)/2)       // upper-triangular tile count (incl. diagonal)

#define LOG2E 1.4426950408889634f
// e^{-c} for c in {0.5, 1, 2, 4}
#define KC0 0.6065306597126334f
#define KC1 0.36787944117144233f
#define KC2 0.1353352832366127f
#define KC3 0.018315638888734179f

__device__ __forceinline__ float fsqrt(float x)  { return __builtin_amdgcn_sqrtf(x); }
__device__ __forceinline__ float frcp(float x)   { return __builtin_amdgcn_rcpf(x); }
__device__ __forceinline__ float fexp2(float x)  { return __builtin_amdgcn_exp2f(x); }
__device__ __forceinline__ float nz(float x)     { return (x == x) ? x : 0.0f; }

// sum of 4 shifted sigmoids: sigma(c-d) = 1/(1 + e^d * e^{-c}); one exp, four rcp
__device__ __forceinline__ float sig4(float dl) {
  float E = fexp2(LOG2E * dl);            // e^{dl}, dl >= 0
  return frcp(1.0f + E * KC0) + frcp(1.0f + E * KC1)
       + frcp(1.0f + E * KC2) + frcp(1.0f + E * KC3);
}

// A (16x4) and B (4x16) f32 WMMA fragments share the same per-lane packing:
// lane l<16  -> token (base + l), components (x, y)
// lane l>=16 -> token (base + l-16), components (z, 0)
__device__ __forceinline__ v2f load_frag(const float* __restrict__ C, int tok, int half) {
  const float* p = C + tok * 3;
  v2f a;
  if (half == 0) { a.x = p[0]; a.y = p[1]; }
  else           { a.x = p[2]; a.y = 0.0f; }
  return a;
}
__device__ __forceinline__ v2f load_frag_gt(const float* __restrict__ C, int tok, int half) {
  const float* p = C + tok * 3;
  v2f a;
  if (half == 0) { a.x = nz(p[0]); a.y = nz(p[1]); }
  else           { a.x = nz(p[2]); a.y = 0.0f; }
  return a;
}

__device__ __forceinline__ v8f wmma4(v2f a, v2f b) {
  v8f c = {};
  // 8 args: (neg_a, A, neg_b, B, c_mod, C, reuse_a, reuse_b)
  return __builtin_amdgcn_wmma_f32_16x16x4_f32(false, a, false, b, (short)0, c, false, false);
}

// -------- prep: packed per-token quantities + MSE partial sums --------------
__global__ void __launch_bounds__(256)
prep_kernel(const float* __restrict__ XL, const float* __restrict__ Xgt,
            const unsigned char* __restrict__ cm,
            const unsigned char* __restrict__ dna,
            const unsigned char* __restrict__ rna,
            const unsigned char* __restrict__ lig,
            const long long* __restrict__ tok,
            float4* __restrict__ rowpack,   // {ngt, mask, cutoff, tok_as_float}
            float4* __restrict__ npredT,    // {np0, np1, np2, np3}
            float* __restrict__ acc)
{
  int l = blockIdx.x * blockDim.x + threadIdx.x;
  if (l >= LTOT) return;

  float gx = nz(Xgt[l*3+0]), gy = nz(Xgt[l*3+1]), gz = nz(Xgt[l*3+2]);
  float ngt = gx*gx + gy*gy + gz*gz;

  float m = cm[l] ? 1.0f : 0.0f;
  int tk = (int)tok[l];
  bool na = (dna[tk] != 0) || (rna[tk] != 0);
  float cut = na ? 30.0f : 15.0f;

  rowpack[l] = make_float4(ngt, m, cut, (float)tk);

  float alpha = (dna[tk] ? 5.0f : 0.0f) + (rna[tk] ? 5.0f : 0.0f) + (lig[tk] ? 10.0f : 0.0f);
  float w = (1.0f + alpha) * m;

  float np[DD];
#pragma unroll
  for (int d = 0; d < DD; ++d) {
    const float* p = XL + (size_t)(d * LTOT + l) * 3;
    float px = p[0], py = p[1], pz = p[2];
    np[d] = px*px + py*py + pz*pz;
    float dx = px - gx, dy = py - gy, dz = pz - gz;
    atomicAdd(&acc[6 + d], w * (dx*dx + dy*dy + dz*dz));
  }
  npredT[l] = make_float4(np[0], np[1], np[2], np[3]);
  atomicAdd(&acc[5], m);
}

// -------- main: WMMA-tiled pairwise lDDT ------------------------------------
__global__ void __launch_bounds__(256)
lddt_tiles(const float* __restrict__ XL, const float* __restrict__ Xgt,
           const float4* __restrict__ rowpack,
           const float4* __restrict__ npredT,
           float* __restrict__ acc)
{
  const int lane = threadIdx.x & 31;
  const int wid  = (blockIdx.x * blockDim.x + threadIdx.x) >> 5;
  const int nw   = (gridDim.x * blockDim.x) >> 5;
  const int tloc = lane & 15;
  const int half = lane >> 4;

  float S[DD] = {0.0f, 0.0f, 0.0f, 0.0f};
  float Nacc = 0.0f;

  for (int k = wid; k < TT; k += nw) {
    // decode linear upper-tri tile index -> (r, c) with c >= r
    float disc = (float)((2*NT+1)*(2*NT+1)) - 8.0f * (float)k;
    int r = (int)(0.5f * ((float)(2*NT+1) - fsqrt(disc)));
    if (r < 0) r = 0;
    while (r > 0 && (r*NT - (r*(r-1))/2) > k) --r;
    while (((r+1)*NT - ((r+1)*r)/2) <= k) ++r;
    int c = r + (k - (r*NT - (r*(r-1))/2));

    const int ibase = r * 16, jbase = c * 16;
    const int j  = jbase + tloc;
    const int ia = ibase + tloc;

    // fragments: gt + 4 samples, rows (A) and cols (B)
    v2f ag = load_frag_gt(Xgt, ia, half);
    v2f bg = load_frag_gt(Xgt, j,  half);
    v2f a_s[DD], b_s[DD];
#pragma unroll
    for (int d = 0; d < DD; ++d) {
      const float* base = XL + (size_t)d * LTOT * 3;
      a_s[d] = load_frag(base, ia, half);
      b_s[d] = load_frag(base, j,  half);
    }

    v8f dg = wmma4(ag, bg);
    v8f dp[DD];
#pragma unroll
    for (int d = 0; d < DD; ++d) dp[d] = wmma4(a_s[d], b_s[d]);

    // per-lane column scalars (2 x b128)
    const float4 rpj = rowpack[j];   // {ngt_j, mask_j, cut_j, tok_j}
    const float4 npj = npredT[j];

    const float wt = (r == c) ? 0.5f : 1.0f;   // diagonal tiles double-count pairs

#pragma unroll
    for (int v = 0; v < 8; ++v) {
      int i = ibase + v + 8 * half;            // C/D layout: m = v (+8 for upper lanes)
      const float4 rpi = rowpack[i];           // 2 x b128 per row
      const float4 npi = npredT[i];

      float gd2 = rpi.x + rpj.x - 2.0f * dg[v];
      float gd  = fsqrt(fmaxf(gd2, 0.0f));
      float cut = (i < j) ? rpi.z : rpj.z;     // cutoff keyed on smaller index (== fi)
      bool pm = (gd > 0.0f) & (gd < cut) & (rpi.y > 0.0f) & (rpj.y > 0.0f) & (rpi.w != rpj.w);
      float pmf = pm ? wt : 0.0f;
      Nacc += pmf;

      const float npi_a[DD] = {npi.x, npi.y, npi.z, npi.w};
      const float npj_a[DD] = {npj.x, npj.y, npj.z, npj.w};
#pragma unroll
      for (int d = 0; d < DD; ++d) {
        float pd = fsqrt(fmaxf(npi_a[d] + npj_a[d] - 2.0f * dp[d][v], 0.0f));
        float dl = fabsf(pd - gd + 1e-6f);
        S[d] += pmf * sig4(dl);
      }
    }
  }

  // wave32 reduction, then one atomic per wave per accumulator
#pragma unroll
  for (int off = 16; off > 0; off >>= 1) {
    Nacc += __shfl_down(Nacc, off, 32);
#pragma unroll
    for (int d = 0; d < DD; ++d) S[d] += __shfl_down(S[d], off, 32);
  }
  if (lane == 0) {
    atomicAdd(&acc[4], Nacc);
#pragma unroll
    for (int d = 0; d < DD; ++d) atomicAdd(&acc[d], S[d]);
  }
}

// -------- finalize ----------------------------------------------------------
__global__ void finalize_kernel(const float* __restrict__ acc,
                                const float* __restrict__ t,
                                float* __restrict__ out)
{
  if (threadIdx.x != 0 || blockIdx.x != 0) return;
  float msum = acc[5];
  float inv3 = 1.0f / (3.0f * (msum + 1e-4f));
  float N = acc[4];
  float sum = 0.0f;
#pragma unroll
  for (int d = 0; d < DD; ++d) {
    float lmse  = acc[6 + d] * inv3;
    float td    = t[d];
    float lam   = (td * td + 256.0f) / ((td * 16.0f) * (td * 16.0f));
    float ldiff = fminf(lam * lmse, 2.0f);
    float lddt  = 0.25f * acc[d] / (N + 1e-6f);
    sum += 0.25f * ldiff + 0.25f * (1.0f - lddt);
  }
  out[0] = 4.0f * sum;   // WEIGHT * (mean(l_diff) + mean(lddt_loss))
}

extern "C" void kernel_launch(void* const* d_in, const int* in_sizes, int n_in,
                              void* d_out, int out_size, void* d_ws, size_t ws_size,
                              hipStream_t stream)
{
  const float*         XL  = (const float*)d_in[0];          // [D,L,3] f32
  const float*         Xgt = (const float*)d_in[1];          // [1,L,3] f32
  const unsigned char* cm  = (const unsigned char*)d_in[2];  // [1,L] bool
  const unsigned char* dna = (const unsigned char*)d_in[3];  // [NTOK] bool
  const unsigned char* rna = (const unsigned char*)d_in[4];  // [NTOK] bool
  const unsigned char* lig = (const unsigned char*)d_in[5];  // [NTOK] bool
  const long long*     tok = (const long long*)d_in[6];      // [L] int64
  const float*         t   = (const float*)d_in[7];          // [D] f32
  float* out = (float*)d_out;

  float*  ws      = (float*)d_ws;
  float4* rowpack = (float4*)ws;                    // L x float4
  float4* npredT  = rowpack + LTOT;                 // L x float4
  float*  acc     = (float*)(npredT + LTOT);        // [0..3]=S_d, [4]=N, [5]=msum, [6..9]=wsq_d

  hipMemsetAsync(acc, 0, 16 * sizeof(float), stream);
  prep_kernel<<<LTOT / 256, 256, 0, stream>>>(XL, Xgt, cm, dna, rna, lig, tok,
                                              rowpack, npredT, acc);
  lddt_tiles<<<1024, 256, 0, stream>>>(XL, Xgt, rowpack, npredT, acc);
  finalize_kernel<<<1, 32, 0, stream>>>(acc, t, out);
}